// DownsamplingBlock_61177514164731
// MI455X (gfx1250) — compile-verified
//
#include <hip/hip_runtime.h>

#define HIN   181
#define WIN   360
#define HOUT  91
#define WOUT  180
#define CIN   128
#define COUT  256
#define KDIM  1152            // CIN * 9
#define NSPAT (HIN * WIN)     // 65160
#define WTILES1 12            // 12 * 32 >= 360
#define NBLK1 (WTILES1 * HIN) // 2172 blocks in kernel 1

typedef float v2f __attribute__((ext_vector_type(2)));
typedef float v8f __attribute__((ext_vector_type(8)));

__device__ __forceinline__ v8f wmma_f32x4(v2f a, v2f b, v8f c) {
  // D = A(16x4 f32) * B(4x16 f32) + C ; full fp32 matrix op on CDNA5
  return __builtin_amdgcn_wmma_f32_16x16x4_f32(false, a, false, b, (short)0, c,
                                               false, false);
}

// LDS strides chosen for bank-conflict-free WMMA fragment reads
#define AS1 25   // Asub row stride (24 cols + 1)
#define XS1 40   // xpsic row stride (32 cols + 8) -> 2*40 mod 64 = 16 (disjoint windows)
#define AS3 18   // skip_w chunk row stride (16 + 2)
#define BS3 72   // Bl row stride (64 + 8)   -> 2*72 mod 64 = 16

// ---------------------------------------------------------------------------
// Kernel 1: fused spectral stencil (xpsi) + main GEMM y = w_conv * xpsi
//   grid = (12, 181), block = 256 (8 waves). Wave tile: 32(M) x 32(N).
// ---------------------------------------------------------------------------
__global__ __launch_bounds__(256)
void k1_spectral_gemm(const float* __restrict__ x,
                      const float* __restrict__ w_conv,
                      const float* __restrict__ psi,
                      const int*   __restrict__ lat_idx,
                      float* __restrict__ yws,
                      float* __restrict__ psum,
                      float* __restrict__ psqs)
{
  __shared__ float xslab[16 * 5 * 36];   // 16 ch x 5 lat rows x (32+4) cols
  __shared__ float xpsic[144 * XS1];     // 144 K-rows x 32 cols (padded)
  __shared__ float Asub [256 * AS1];     // 256 M-rows x 24 K-cols (padded)
  __shared__ float psis [9 * 25];        // psi[:, h, :, :]

  const int t     = threadIdx.x;
  const int h     = blockIdx.y;
  const int w0    = blockIdx.x * 32;
  const int lane  = t & 31;
  const int wave  = t >> 5;
  const int laneN = lane & 15;
  const int hi    = lane >> 4;
  const int mbase = wave * 32;

  if (t < 225) {                         // stage psi row for this h
    int kk = t / 25, pq = t - kk * 25;
    psis[t] = psi[(kk * HIN + h) * 25 + pq];
  }
  int rows[5];
  #pragma unroll
  for (int p = 0; p < 5; ++p) rows[p] = lat_idx[h * 5 + p];

  const v8f vzero = {};
  v8f acc[2][2];
  #pragma unroll
  for (int mt = 0; mt < 2; ++mt)
    #pragma unroll
    for (int nt = 0; nt < 2; ++nt) acc[mt][nt] = vzero;

  for (int cc = 0; cc < CIN; cc += 16) {
    __syncthreads();
    // ---- stage x halo slab for 16 channels (circular in w) ----
    for (int i = t; i < 16 * 5 * 36; i += 256) {
      int c   = i / 180;
      int rp  = i - c * 180;
      int p   = rp / 36;
      int col = rp - p * 36;
      int w   = w0 + col - 2;
      if (w < 0)     w += WIN;
      if (w >= WIN)  w -= WIN;
      xslab[i] = x[((cc + c) * HIN + rows[p]) * WIN + w];
    }
    __syncthreads();
    // ---- build xpsi chunk: 144 K-rows x 32 pixels, 25 MACs each ----
    for (int i = t; i < 144 * 32; i += 256) {
      int n  = i & 31;
      int jl = i >> 5;            // 0..143 = local (c,k)
      int c  = jl / 9;
      int kk = jl - c * 9;
      const float* ps = &psis[kk * 25];
      const float* xs = &xslab[c * 180];
      float s = 0.0f;
      #pragma unroll
      for (int p = 0; p < 5; ++p)
        #pragma unroll
        for (int q = 0; q < 5; ++q)
          s += xs[p * 36 + n + q] * ps[p * 5 + q];
      xpsic[jl * XS1 + n] = s;
    }
    __syncthreads();
    // ---- GEMM over this chunk's 144 K values (6 x 24-col A stages) ----
    for (int ks = 0; ks < 144; ks += 24) {
      for (int i = t; i < 256 * 24; i += 256) {
        int m  = i / 24;
        int kl = i - m * 24;
        Asub[m * AS1 + kl] = w_conv[m * KDIM + cc * 9 + ks + kl];
      }
      __syncthreads();
      #pragma unroll
      for (int kk = 0; kk < 6; ++kk) {
        const int k0 = kk * 4;
        v2f a[2], b[2];
        #pragma unroll
        for (int mt = 0; mt < 2; ++mt) {
          const int m = mbase + mt * 16 + laneN;
          a[mt].x = Asub[m * AS1 + k0 + 2 * hi];
          a[mt].y = Asub[m * AS1 + k0 + 2 * hi + 1];
        }
        #pragma unroll
        for (int nt = 0; nt < 2; ++nt) {
          const int jr = ks + k0 + 2 * hi;
          b[nt].x = xpsic[jr * XS1 + nt * 16 + laneN];
          b[nt].y = xpsic[(jr + 1) * XS1 + nt * 16 + laneN];
        }
        #pragma unroll
        for (int mt = 0; mt < 2; ++mt)
          #pragma unroll
          for (int nt = 0; nt < 2; ++nt)
            acc[mt][nt] = wmma_f32x4(a[mt], b[nt], acc[mt][nt]);
      }
      __syncthreads();
    }
  }

  // ---- epilogue: store y (coalesced per half-wave) + BN partial sums ----
  const int b1 = blockIdx.y * WTILES1 + blockIdx.x;
  #pragma unroll
  for (int mt = 0; mt < 2; ++mt) {
    #pragma unroll
    for (int r = 0; r < 8; ++r) {
      const int o = mbase + mt * 16 + r + 8 * hi;
      float sv = 0.0f, sq = 0.0f;
      #pragma unroll
      for (int nt = 0; nt < 2; ++nt) {
        const int w = w0 + nt * 16 + laneN;
        const float v = acc[mt][nt][r];
        if (w < WIN) {
          yws[o * NSPAT + h * WIN + w] = v;
          sv += v;
          sq += v * v;
        }
      }
      #pragma unroll
      for (int m = 8; m >= 1; m >>= 1) {   // reduce over 16-lane half-wave
        sv += __shfl_xor(sv, m, 32);
        sq += __shfl_xor(sq, m, 32);
      }
      if (laneN == 0) {                    // deterministic per-block partials
        psum[b1 * COUT + o] = sv;
        psqs[b1 * COUT + o] = sq;
      }
    }
  }
}

// ---------------------------------------------------------------------------
// Kernel 2: reduce BN partials -> scale/shift per channel
// ---------------------------------------------------------------------------
__global__ __launch_bounds__(256)
void k2_bn_finalize(const float* __restrict__ psum,
                    const float* __restrict__ psqs,
                    const float* __restrict__ gamma,
                    const float* __restrict__ beta,
                    float* __restrict__ scale,
                    float* __restrict__ shift)
{
  const int o = threadIdx.x;
  float s = 0.0f, q = 0.0f;
  for (int b = 0; b < NBLK1; ++b) {       // deterministic order
    s += psum[b * COUT + o];
    q += psqs[b * COUT + o];
  }
  const float inv_n = 1.0f / (float)NSPAT;
  const float mean  = s * inv_n;
  const float var   = q * inv_n - mean * mean;
  const float sc    = gamma[o] * rsqrtf(var + 1e-5f);
  scale[o] = sc;
  shift[o] = beta[o] - mean * sc;
}

// ---------------------------------------------------------------------------
// Kernel 3: skip GEMM (WMMA) at the gathered bilinear points + BN/ReLU + interp
//   grid = (12, 91), block = 256. n = corner*16 + wo_local (4 n-tiles/wave).
// ---------------------------------------------------------------------------
__global__ __launch_bounds__(256)
void k3_skip_interp(const float* __restrict__ x,
                    const float* __restrict__ skip_w,
                    const float* __restrict__ skip_b,
                    const float* __restrict__ yws,
                    const float* __restrict__ scale,
                    const float* __restrict__ shift,
                    const int*   __restrict__ lat_lo,
                    const float* __restrict__ lat_w,
                    const int*   __restrict__ lon_lo,
                    const float* __restrict__ lon_w,
                    float* __restrict__ out)
{
  __shared__ float Bl  [128 * BS3];   // x at 64 gathered points (K=128 rows)
  __shared__ float Asub[256 * AS3];   // skip_w chunk: 256 x 16

  const int t     = threadIdx.x;
  const int ho    = blockIdx.y;
  const int wo0   = blockIdx.x * 16;
  const int lane  = t & 31;
  const int wave  = t >> 5;
  const int laneN = lane & 15;
  const int hi    = lane >> 4;
  const int mbase = wave * 32;

  const int   r0 = lat_lo[ho];
  const float aw = lat_w[ho];

  // ---- gather B: 4 bilinear corners x 16 output columns ----
  for (int i = t; i < 128 * 64; i += 256) {
    int c      = i >> 6;
    int n      = i & 63;
    int corner = n >> 4;
    int wl     = n & 15;
    int wo     = wo0 + wl; if (wo > WOUT - 1) wo = WOUT - 1;
    int col    = lon_lo[wo] + (corner & 1);
    if (col >= WIN) col -= WIN;
    int row    = r0 + (corner >> 1);
    Bl[c * BS3 + n] = x[(c * HIN + row) * WIN + col];
  }

  const v8f vzero = {};
  v8f acc[2][4];
  #pragma unroll
  for (int mt = 0; mt < 2; ++mt)
    #pragma unroll
    for (int nt = 0; nt < 4; ++nt) acc[mt][nt] = vzero;
  __syncthreads();

  for (int kc = 0; kc < CIN; kc += 16) {
    for (int i = t; i < 256 * 16; i += 256) {
      int m  = i >> 4;
      int kl = i & 15;
      Asub[m * AS3 + kl] = skip_w[m * CIN + kc + kl];
    }
    __syncthreads();
    #pragma unroll
    for (int kk = 0; kk < 4; ++kk) {
      const int k0 = kk * 4;
      v2f a[2], b[4];
      #pragma unroll
      for (int mt = 0; mt < 2; ++mt) {
        const int m = mbase + mt * 16 + laneN;
        a[mt].x = Asub[m * AS3 + k0 + 2 * hi];
        a[mt].y = Asub[m * AS3 + k0 + 2 * hi + 1];
      }
      #pragma unroll
      for (int nt = 0; nt < 4; ++nt) {
        const int kr = kc + k0 + 2 * hi;
        b[nt].x = Bl[kr * BS3 + nt * 16 + laneN];
        b[nt].y = Bl[(kr + 1) * BS3 + nt * 16 + laneN];
      }
      #pragma unroll
      for (int mt = 0; mt < 2; ++mt)
        #pragma unroll
        for (int nt = 0; nt < 4; ++nt)
          acc[mt][nt] = wmma_f32x4(a[mt], b[nt], acc[mt][nt]);
    }
    __syncthreads();
  }

  // ---- epilogue: z = skip + b + relu(y*scale+shift); bilinear combine ----
  const int  wo    = wo0 + laneN;
  const bool valid = wo < WOUT;
  const int  woc   = valid ? wo : (WOUT - 1);
  const float lw   = lon_w[woc];
  int c0 = lon_lo[woc];
  int c1 = c0 + 1; if (c1 >= WIN) c1 -= WIN;
  const int r1 = r0 + 1;

  #pragma unroll
  for (int mt = 0; mt < 2; ++mt) {
    #pragma unroll
    for (int r = 0; r < 8; ++r) {
      const int o  = mbase + mt * 16 + r + 8 * hi;
      const float sb = skip_b[o];
      const float sc = scale[o];
      const float sh = shift[o];
      const int yb = o * NSPAT;
      const float z00 = acc[mt][0][r] + sb + fmaxf(yws[yb + r0 * WIN + c0] * sc + sh, 0.0f);
      const float z01 = acc[mt][1][r] + sb + fmaxf(yws[yb + r0 * WIN + c1] * sc + sh, 0.0f);
      const float z10 = acc[mt][2][r] + sb + fmaxf(yws[yb + r1 * WIN + c0] * sc + sh, 0.0f);
      const float z11 = acc[mt][3][r] + sb + fmaxf(yws[yb + r1 * WIN + c1] * sc + sh, 0.0f);
      const float zl0 = z00 + lw * (z01 - z00);
      const float zl1 = z10 + lw * (z11 - z10);
      const float res = zl0 + aw * (zl1 - zl0);
      if (valid) out[(o * HOUT + ho) * WOUT + wo] = res;
    }
  }
}

// ---------------------------------------------------------------------------
extern "C" void kernel_launch(void* const* d_in, const int* in_sizes, int n_in,
                              void* d_out, int out_size, void* d_ws, size_t ws_size,
                              hipStream_t stream) {
  (void)in_sizes; (void)n_in; (void)out_size; (void)ws_size;
  const float* x       = (const float*)d_in[0];
  const float* w_conv  = (const float*)d_in[1];
  const float* gamma   = (const float*)d_in[2];
  const float* beta    = (const float*)d_in[3];
  const float* skip_w  = (const float*)d_in[4];
  const float* skip_b  = (const float*)d_in[5];
  const float* psi     = (const float*)d_in[6];
  const int*   lat_idx = (const int*)d_in[7];
  const int*   lat_lo  = (const int*)d_in[8];
  const float* lat_w   = (const float*)d_in[9];
  const int*   lon_lo  = (const int*)d_in[10];
  const float* lon_w   = (const float*)d_in[11];
  float* out = (float*)d_out;

  float* yws   = (float*)d_ws;                       // COUT*NSPAT
  float* psum  = yws  + (size_t)COUT * NSPAT;        // NBLK1*COUT
  float* psqs  = psum + (size_t)NBLK1 * COUT;        // NBLK1*COUT
  float* scale = psqs + (size_t)NBLK1 * COUT;        // COUT
  float* shift = scale + COUT;                       // COUT

  k1_spectral_gemm<<<dim3(WTILES1, HIN), 256, 0, stream>>>(
      x, w_conv, psi, lat_idx, yws, psum, psqs);
  k2_bn_finalize<<<1, COUT, 0, stream>>>(psum, psqs, gamma, beta, scale, shift);
  k3_skip_interp<<<dim3(12, HOUT), 256, 0, stream>>>(
      x, skip_w, skip_b, yws, scale, shift, lat_lo, lat_w, lon_lo, lon_w, out);
}